// Attention_66735201846029
// MI455X (gfx1250) — compile-verified
//
#include <hip/hip_runtime.h>
#include <hip/hip_bf16.h>
#include <math.h>

typedef float v2f __attribute__((ext_vector_type(2)));
typedef float v8f __attribute__((ext_vector_type(8)));

#define BDIM 32
#define SDIM 4096
#define HDIM 1024
#define WIN  10

// ---------------------------------------------------------------- zero ctx
__global__ void zero_f32(float* __restrict__ p, int n) {
  int i = blockIdx.x * blockDim.x + threadIdx.x;
  if (i < n) p[i] = 0.0f;
}

// ---------------------------------------------------------------- proj = dec @ W^T  (WMMA f32 16x16x4)
// proj[b,h] = sum_k dec[b,k] * W[h,k]
// One wave (32 threads) owns a 16x16 output tile; K-loop steps of 4.
// A (16x4, 32-bit): lanes 0-15 -> M=lane, V0=K0,V1=K1 ; lanes 16-31 -> V0=K2,V1=K3
// B (4x16, 32-bit): lanes 0-15 -> N=lane, V0=K0,V1=K1 ; lanes 16-31 -> V0=K2,V1=K3
// C/D (16x16 f32): VGPR r -> M = r + 8*(lane>=16), N = lane&15
__global__ void __launch_bounds__(32)
proj_gemm_wmma(const float* __restrict__ dec,
               const float* __restrict__ W,
               float* __restrict__ proj) {
  const int lane = threadIdx.x;
  const int l    = lane & 15;
  const int half = lane >> 4;          // 0: K0/K1 , 1: K2/K3
  const int n0   = blockIdx.x << 4;    // h tile
  const int m0   = blockIdx.y << 4;    // b tile

  const float* __restrict__ arow = dec + (m0 + l) * HDIM + 2 * half;  // dec[b, k]
  const float* __restrict__ brow = W   + (n0 + l) * HDIM + 2 * half;  // W[h, k] == W^T[k, h]

  v8f acc = {};
  #pragma unroll 4
  for (int k0 = 0; k0 < HDIM; k0 += 4) {
    v2f a, b;
    a.x = arow[k0];  a.y = arow[k0 + 1];
    b.x = brow[k0];  b.y = brow[k0 + 1];
    acc = __builtin_amdgcn_wmma_f32_16x16x4_f32(
        /*neg_a=*/false, a, /*neg_b=*/false, b,
        /*c_mod=*/(short)0, acc, /*reuse_a=*/false, /*reuse_b=*/false);
  }
  #pragma unroll
  for (int r = 0; r < 8; ++r) {
    proj[(m0 + r + 8 * half) * HDIM + (n0 + l)] = acc[r];
  }
}

// ---------------------------------------------------------------- energy[b,s] = enc[b,s,:] . proj[b,:]
// One wave per row; float4 streaming loads of enc (512 MB pass #1).
__global__ void __launch_bounds__(256)
energy_dot(const float* __restrict__ enc,
           const float* __restrict__ proj,
           float* __restrict__ energy) {
  const int lane = threadIdx.x & 31;
  const int wv   = threadIdx.x >> 5;
  const size_t row = (size_t)blockIdx.x * 8 + wv;      // b*S + s
  const int b = (int)(row >> 12);                      // /4096

  const float4* __restrict__ e4 = (const float4*)(enc + row * HDIM);
  const float4* __restrict__ p4 = (const float4*)(proj + (size_t)b * HDIM);

  float sum = 0.0f;
  #pragma unroll
  for (int i = lane; i < HDIM / 4; i += 32) {
    float4 x = e4[i];
    float4 y = p4[i];
    sum = fmaf(x.x, y.x, sum);
    sum = fmaf(x.y, y.y, sum);
    sum = fmaf(x.z, y.z, sum);
    sum = fmaf(x.w, y.w, sum);
  }
  #pragma unroll
  for (int off = 16; off; off >>= 1) sum += __shfl_down(sum, off, 32);
  if (lane == 0) energy[row] = sum;
}

// ---------------------------------------------------------------- masked softmax over S (multiplicative mask, faithful)
__global__ void __launch_bounds__(256)
masked_softmax(const float* __restrict__ energy,
               const int* __restrict__ pos_ptr,
               float* __restrict__ attn) {
  const int b    = blockIdx.x;
  const int t    = threadIdx.x;
  const int lane = t & 31;
  const int wv   = t >> 5;

  const int pos = *pos_ptr;
  int lo = pos - WIN;     if (lo < 0) lo = 0;
  int hi = pos + WIN - 1; if (hi > SDIM - 1) hi = SDIM - 1;

  const float* __restrict__ erow = energy + (size_t)b * SDIM;

  float vals[SDIM / 256];
  float mx = -INFINITY;
  #pragma unroll
  for (int i = 0; i < SDIM / 256; ++i) {
    int s = t + 256 * i;
    float e = erow[s];
    // align_wts * energy : 1.0 inside window, -1e10 outside (multiplicative!)
    float v = ((s >= lo) && (s <= hi)) ? e : (-1e10f * e);
    vals[i] = v;
    mx = fmaxf(mx, v);
  }

  __shared__ float red[8];
  #pragma unroll
  for (int off = 16; off; off >>= 1) mx = fmaxf(mx, __shfl_down(mx, off, 32));
  if (lane == 0) red[wv] = mx;
  __syncthreads();
  float bm = fmaxf(fmaxf(fmaxf(red[0], red[1]), fmaxf(red[2], red[3])),
                   fmaxf(fmaxf(red[4], red[5]), fmaxf(red[6], red[7])));
  __syncthreads();

  float sum = 0.0f;
  #pragma unroll
  for (int i = 0; i < SDIM / 256; ++i) {
    float ex = __expf(vals[i] - bm);
    vals[i] = ex;
    sum += ex;
  }
  #pragma unroll
  for (int off = 16; off; off >>= 1) sum += __shfl_down(sum, off, 32);
  if (lane == 0) red[wv] = sum;
  __syncthreads();
  float bs = ((red[0] + red[1]) + (red[2] + red[3])) +
             ((red[4] + red[5]) + (red[6] + red[7]));
  float inv = 1.0f / bs;

  float* __restrict__ arow = attn + (size_t)b * SDIM;
  #pragma unroll
  for (int i = 0; i < SDIM / 256; ++i) {
    arow[t + 256 * i] = vals[i] * inv;
  }
}

// ---------------------------------------------------------------- context[b,h] = sum_s attn[b,s] * enc[b,s,h]
// (b, h-chunk of 256, s-segment of 512) per block; 1024 blocks streaming enc (pass #2).
__global__ void __launch_bounds__(256)
context_accum(const float* __restrict__ enc,
              const float* __restrict__ attn,
              float* __restrict__ ctx) {
  const int b  = blockIdx.x;
  const int h  = (blockIdx.y << 8) + threadIdx.x;
  const int s0 = blockIdx.z << 9;

  const float* __restrict__ base = enc + ((size_t)b * SDIM + s0) * HDIM + h;
  const float* __restrict__ aw   = attn + (size_t)b * SDIM + s0;

  float acc = 0.0f;
  #pragma unroll 4
  for (int s = 0; s < 512; ++s) {
    acc = fmaf(aw[s], base[(size_t)s * HDIM], acc);
  }
  atomicAdd(&ctx[(size_t)b * HDIM + h], acc);
}

// ---------------------------------------------------------------- launch
extern "C" void kernel_launch(void* const* d_in, const int* in_sizes, int n_in,
                              void* d_out, int out_size, void* d_ws, size_t ws_size,
                              hipStream_t stream) {
  (void)in_sizes; (void)n_in; (void)out_size; (void)ws_size;

  const float* dec = (const float*)d_in[0];   // [32,1024]
  const float* enc = (const float*)d_in[1];   // [32,4096,1024]
  const float* W   = (const float*)d_in[2];   // [1024,1024]
  const int*   pos = (const int*)d_in[3];     // scalar

  float* ctx_out  = (float*)d_out;                       // [32,1,1024]
  float* attn_out = (float*)d_out + BDIM * HDIM;         // [32,4096,1]

  float* proj   = (float*)d_ws;                          // 32*1024 f32
  float* energy = proj + BDIM * HDIM;                    // 32*4096 f32

  zero_f32<<<(BDIM * HDIM + 255) / 256, 256, 0, stream>>>(ctx_out, BDIM * HDIM);
  proj_gemm_wmma<<<dim3(HDIM / 16, BDIM / 16), 32, 0, stream>>>(dec, W, proj);
  energy_dot<<<(BDIM * SDIM) / 8, 256, 0, stream>>>(enc, proj, energy);
  masked_softmax<<<BDIM, 256, 0, stream>>>(energy, pos, attn_out);
  context_accum<<<dim3(BDIM, HDIM / 256, SDIM / 512), 256, 0, stream>>>(enc, attn_out, ctx_out);
}